// YoloV3NonMaxSuppression_69243462746567
// MI455X (gfx1250) — compile-verified
//
#include <hip/hip_runtime.h>
#include <hip/hip_bf16.h>

#define NPRED 22743
#define NB 16
#define NCLS 80
#define PSTRIDE 85
#define KTOP 512
#define NPT 23            // ceil(22743 / 1024)
#define MIN_WH_F 2.0f
#define MAX_WH_F 10000.0f
#define CONF_THRES_F 0.001f
#define NMS_THRES_F 0.5f

typedef __attribute__((ext_vector_type(2))) float v2f;
typedef __attribute__((ext_vector_type(8))) float v8f;
typedef int v4i_vs __attribute__((vector_size(16)));   // matches builtin's pointee

#if __has_builtin(__builtin_amdgcn_wmma_f32_16x16x4_f32)
#define HAVE_WMMA_F32X4 1
#else
#define HAVE_WMMA_F32X4 0
#endif

// gfx1250 async global->LDS path (ASYNCcnt-tracked). Guarded: falls back to
// plain loads if the toolchain doesn't expose the builtins.
#if __has_builtin(__builtin_amdgcn_global_load_async_to_lds_b32) && \
    __has_builtin(__builtin_amdgcn_global_load_async_to_lds_b128) && \
    __has_builtin(__builtin_amdgcn_s_wait_asynccnt)
#define HAVE_ASYNC_LDS 1
#else
#define HAVE_ASYNC_LDS 0
#endif

#if HAVE_ASYNC_LDS
// Generic->AS(1): global addresses are identical in flat and global spaces.
// Generic->AS(3): LDS byte offset is the low 32 bits of the flat shared addr.
__device__ __forceinline__ void async_ld_b32(const void* g, void* l) {
    __builtin_amdgcn_global_load_async_to_lds_b32(
        (__attribute__((address_space(1))) int*)(unsigned long long)g,
        (__attribute__((address_space(3))) int*)(unsigned)(unsigned long long)l,
        0, 0);
}
__device__ __forceinline__ void async_ld_b128(const void* g, void* l) {
    __builtin_amdgcn_global_load_async_to_lds_b128(
        (__attribute__((address_space(1))) v4i_vs*)(unsigned long long)g,
        (__attribute__((address_space(3))) v4i_vs*)(unsigned)(unsigned long long)l,
        0, 0);
}
#endif

// ---------------------------------------------------------------------------
// Kernel 1: per-image scoring + exact top-512 selection + gather.
// One 1024-thread block (32 wave32 waves) per image. Each lane keeps 23 masked
// confidences in VGPRs; 512 rounds of block argmax with (val desc, idx asc)
// tie-break reproduce lax.top_k exactly. Then threads 0..511 gather the
// winners' boxes / conf / class and build the class-offset boxes + areas.
// ---------------------------------------------------------------------------
__global__ __launch_bounds__(1024) void yolo_topk_gather(
    const float* __restrict__ pred,
    float* __restrict__ conf_k, float* __restrict__ cc_k,
    float* __restrict__ cp_k, float* __restrict__ a1,
    int* __restrict__ ok_k, float* __restrict__ obox)
{
    const int img  = blockIdx.x;
    const int t    = threadIdx.x;
    const int lane = t & 31;
    const int wv   = t >> 5;
    const float* P = pred + (size_t)img * NPRED * PSTRIDE;

    float lc[NPT];
    #pragma unroll
    for (int s = 0; s < NPT; ++s) {
        const int p = t + (s << 10);
        float cm = -3.0f;                      // padding: below masked (-1)
        if (p < NPRED) {
            const float* r = P + (size_t)p * PSTRIDE;
            const float cx = r[0], cy = r[1], w = r[2], h = r[3], obj = r[4];
            bool fin = isfinite(cx) && isfinite(cy) && isfinite(w) &&
                       isfinite(h) && isfinite(obj);
            float best = r[5];
            fin = fin && isfinite(best);
            for (int c = 1; c < NCLS; ++c) {
                const float v = r[5 + c];
                fin = fin && isfinite(v);
                if (v > best) best = v;        // first-max tie-break (argmax)
            }
            const float conf = obj * best;
            const bool ok = (conf > CONF_THRES_F) &&
                            (w > MIN_WH_F) && (w < MAX_WH_F) &&
                            (h > MIN_WH_F) && (h < MAX_WH_F) && fin;
            cm = ok ? conf : -1.0f;
        }
        lc[s] = cm;
    }

    __shared__ float s_val[32];
    __shared__ int   s_idx[32];
    __shared__ int   s_win;
    __shared__ int   s_topk[KTOP];

    for (int k = 0; k < KTOP; ++k) {
        float bv = -4.0f;
        int   bi = 0x7fffffff;
        #pragma unroll
        for (int s = 0; s < NPT; ++s) {
            if (lc[s] > bv) { bv = lc[s]; bi = t + (s << 10); }
        }
        for (int off = 16; off; off >>= 1) {
            const float ov = __shfl_down(bv, off);
            const int   oi = __shfl_down(bi, off);
            if (ov > bv || (ov == bv && oi < bi)) { bv = ov; bi = oi; }
        }
        if (lane == 0) { s_val[wv] = bv; s_idx[wv] = bi; }
        __syncthreads();
        if (wv == 0) {
            float v2 = s_val[lane];
            int   i2 = s_idx[lane];
            for (int off = 16; off; off >>= 1) {
                const float ov = __shfl_down(v2, off);
                const int   oi = __shfl_down(i2, off);
                if (ov > v2 || (ov == v2 && oi < i2)) { v2 = ov; i2 = oi; }
            }
            if (lane == 0) { s_win = i2; s_topk[k] = i2; }
        }
        __syncthreads();
        const int widx = s_win;
        if ((widx & 1023) == t) lc[widx >> 10] = -3.0f;   // remove winner
    }
    __syncthreads();

    if (t < KTOP) {
        const int idx  = s_topk[t];
        const float* r = P + (size_t)idx * PSTRIDE;
        const float cx = r[0], cy = r[1], w = r[2], h = r[3], obj = r[4];
        bool fin = isfinite(cx) && isfinite(cy) && isfinite(w) &&
                   isfinite(h) && isfinite(obj);
        float best = r[5];
        int   bc   = 0;
        fin = fin && isfinite(best);
        for (int c = 1; c < NCLS; ++c) {
            const float v = r[5 + c];
            fin = fin && isfinite(v);
            if (v > best) { best = v; bc = c; }
        }
        const float conf = obj * best;
        const bool ok = (conf > CONF_THRES_F) &&
                        (w > MIN_WH_F) && (w < MAX_WH_F) &&
                        (h > MIN_WH_F) && (h < MAX_WH_F) && fin;
        const float x0 = cx - w * 0.5f, y0 = cy - h * 0.5f;
        const float x1 = cx + w * 0.5f, y1 = cy + h * 0.5f;
        const float off = (float)bc * MAX_WH_F;
        const float ox0 = x0 + off, ox1 = x1 + off;
        const size_t o = (size_t)img * KTOP + t;
        conf_k[o] = conf;
        cc_k[o]   = best;
        cp_k[o]   = (float)bc;
        ok_k[o]   = ok ? 1 : 0;
        obox[o * 4 + 0] = ox0;
        obox[o * 4 + 1] = y0;
        obox[o * 4 + 2] = ox1;
        obox[o * 4 + 3] = y1;
        a1[o] = (ox1 - ox0) * (y1 - y0);
    }
}

// ---------------------------------------------------------------------------
// Kernel 2: per-image 512x512 IoU matrix, hoisted out of the serial scan.
// Denominator outer-sum a1[i] + a2[j] is a rank-2 GEMM computed exactly in
// f32 by V_WMMA_F32_16X16X4_F32 (one WMMA per 16x16 tile); intersections in
// VALU from LDS-cached boxes (staged by the gfx1250 async global->LDS path).
// 16 MB result is L2-resident (192 MB L2). One wave per tile; 8 waves per
// block; 128 blocks per image.
// ---------------------------------------------------------------------------
__global__ __launch_bounds__(256) void yolo_iou_matrix(
    const float* __restrict__ obox, const float* __restrict__ a1,
    float* __restrict__ iou)
{
    const int img  = blockIdx.y;
    const int t    = threadIdx.x;
    const int lane = t & 31;
    const int wv   = t >> 5;

    __shared__ __align__(16) float s_ob[KTOP * 4];
    __shared__ __align__(16) float s_a[KTOP];
    const size_t base = (size_t)img * KTOP;
#if HAVE_ASYNC_LDS
    async_ld_b128(obox + base * 4 + (size_t)t * 8,     &s_ob[t * 8]);
    async_ld_b128(obox + base * 4 + (size_t)t * 8 + 4, &s_ob[t * 8 + 4]);
    async_ld_b32(a1 + base + t,       &s_a[t]);
    async_ld_b32(a1 + base + 256 + t, &s_a[256 + t]);
    __builtin_amdgcn_s_wait_asynccnt(0);
#else
    for (int c = t; c < KTOP * 4; c += 256) s_ob[c] = obox[base * 4 + c];
    for (int c = t; c < KTOP; c += 256)     s_a[c]  = a1[base + c];
#endif
    __syncthreads();

    const int tile = blockIdx.x * 8 + wv;       // 0..1023
    const int i0   = (tile >> 5) << 4;
    const int j0   = (tile & 31) << 4;

#if HAVE_WMMA_F32X4
    // A (16x4): VGPR0 = K0 (lanes 0-15) / K2 (lanes 16-31); VGPR1 = K1 / K3.
    // A = [a1 | 1 | 0 | 0], B = [1 ; a2 ; 0 ; 0]  =>  D[m][n] = a1[i0+m]+a2[j0+n]
    v2f A, Bm;
    A.x  = (lane < 16) ? s_a[i0 + lane] : 0.0f;   // K=0 row / K=2 row(=0)
    A.y  = (lane < 16) ? 1.0f           : 0.0f;   // K=1 row / K=3 row(=0)
    Bm.x = (lane < 16) ? 1.0f           : 0.0f;   // K=0 row / K=2 row(=0)
    Bm.y = (lane < 16) ? s_a[j0 + lane] : 0.0f;   // K=1 row / K=3 row(=0)
    v8f Cz = {};
    v8f d = __builtin_amdgcn_wmma_f32_16x16x4_f32(
        false, A, false, Bm, (short)0, Cz, false, false);
#endif

    // D layout: VGPR r, lanes 0-15 -> (M=r, N=lane); lanes 16-31 -> (M=r+8, N=lane-16)
    const int mhi = (lane >= 16) ? 8 : 0;
    const int n   = lane & 15;
    #pragma unroll
    for (int r = 0; r < 8; ++r) {
        const int i = i0 + r + mhi;
        const int j = j0 + n;
        const float bx0 = s_ob[i * 4 + 0], by0 = s_ob[i * 4 + 1];
        const float bx1 = s_ob[i * 4 + 2], by1 = s_ob[i * 4 + 3];
        const float cx0 = s_ob[j * 4 + 0], cy0 = s_ob[j * 4 + 1];
        const float cx1 = s_ob[j * 4 + 2], cy1 = s_ob[j * 4 + 3];
        float ix = fminf(bx1, cx1) - fmaxf(bx0, cx0); ix = fmaxf(ix, 0.0f);
        float iy = fminf(by1, cy1) - fmaxf(by0, cy0); iy = fmaxf(iy, 0.0f);
        const float inter = ix * iy;
#if HAVE_WMMA_F32X4
        const float asum = d[r];
#else
        const float asum = s_a[i] + s_a[j];
#endif
        iou[(base + i) * KTOP + j] = inter / (asum - inter + 1e-9f);
    }
}

// ---------------------------------------------------------------------------
// Kernel 3: the serial 512-step NMS scan with weighted merging. One
// 512-thread block per image. The IoU row for step i+1 is prefetched into a
// double-buffered LDS row by the gfx1250 async global->LDS engine while step
// i runs its reduction, hiding L2 latency on the serial chain. Each thread
// only reads the element it issued, so a per-wave s_wait_asynccnt(0) is the
// only ordering needed for the row buffer.
// ---------------------------------------------------------------------------
__global__ __launch_bounds__(512) void yolo_nms_scan(
    const float* __restrict__ obox, const float* __restrict__ conf_k,
    const int* __restrict__ ok_k, const float* __restrict__ iou,
    int* __restrict__ keep, float* __restrict__ merged)
{
    const int img  = blockIdx.x;
    const int t    = threadIdx.x;
    const int lane = t & 31;
    const int wv   = t >> 5;

    __shared__ __align__(16) float s_ob[KTOP * 4];
    __shared__ float s_cf[KTOP];
    __shared__ int   s_alive[KTOP];
    __shared__ float s_part[16][5];
    __shared__ float s_m[4];

    const size_t base = (size_t)img * KTOP;
    const float* R = iou + base * KTOP;

#if HAVE_ASYNC_LDS
    __shared__ float s_row[2][KTOP];
    async_ld_b128(obox + (base + t) * 4, &s_ob[t * 4]);
    async_ld_b32(conf_k + base + t, &s_cf[t]);
    async_ld_b32(ok_k + base + t, &s_alive[t]);
    async_ld_b32(R + t, &s_row[0][t]);           // prefetch row 0
    __builtin_amdgcn_s_wait_asynccnt(0);
#else
    #pragma unroll
    for (int c = 0; c < 4; ++c) s_ob[t * 4 + c] = obox[(base + t) * 4 + c];
    s_cf[t]    = conf_k[base + t];
    s_alive[t] = ok_k[base + t];
#endif
    __syncthreads();

    const float b0 = s_ob[t * 4 + 0], b1 = s_ob[t * 4 + 1];
    const float b2 = s_ob[t * 4 + 2], b3 = s_ob[t * 4 + 3];
    const float cf = s_cf[t];

    for (int i = 0; i < KTOP; ++i) {
#if HAVE_ASYNC_LDS
        __builtin_amdgcn_s_wait_asynccnt(0);
        const float u = s_row[i & 1][t];
        if (i + 1 < KTOP)
            async_ld_b32(R + (size_t)(i + 1) * KTOP + t, &s_row[(i + 1) & 1][t]);
#else
        const float u = R[(size_t)i * KTOP + t];
#endif
        const int  alive_t = s_alive[t];
        const int  take    = s_alive[i];
        const int  sel     = alive_t && (u > NMS_THRES_F);
        const float wgt    = sel ? cf : 0.0f;
        float p0 = wgt * b0, p1 = wgt * b1, p2 = wgt * b2, p3 = wgt * b3, pw = wgt;

        for (int off = 16; off; off >>= 1) {
            p0 += __shfl_down(p0, off);
            p1 += __shfl_down(p1, off);
            p2 += __shfl_down(p2, off);
            p3 += __shfl_down(p3, off);
            pw += __shfl_down(pw, off);
        }
        if (lane == 0) {
            s_part[wv][0] = p0; s_part[wv][1] = p1; s_part[wv][2] = p2;
            s_part[wv][3] = p3; s_part[wv][4] = pw;
        }
        __syncthreads();
        if (wv == 0) {
            float q0 = 0.f, q1 = 0.f, q2 = 0.f, q3 = 0.f, qw = 0.f;
            if (lane < 16) {
                q0 = s_part[lane][0]; q1 = s_part[lane][1]; q2 = s_part[lane][2];
                q3 = s_part[lane][3]; qw = s_part[lane][4];
            }
            for (int off = 8; off; off >>= 1) {
                q0 += __shfl_down(q0, off);
                q1 += __shfl_down(q1, off);
                q2 += __shfl_down(q2, off);
                q3 += __shfl_down(q3, off);
                qw += __shfl_down(qw, off);
            }
            if (lane == 0) {
                const float W = qw + 1e-12f;
                s_m[0] = q0 / W; s_m[1] = q1 / W; s_m[2] = q2 / W; s_m[3] = q3 / W;
            }
        }
        __syncthreads();
        if (take) {
            if (t < 4) merged[(base + i) * 4 + t] = s_m[t];
            if (sel) s_alive[t] = 0;
        }
        if (t == i) keep[base + i] = take;
        __syncthreads();
    }
}

// ---------------------------------------------------------------------------
// Kernel 4: stable re-sort of the 512 slots by kept-conf (rank-by-count,
// ties -> lower index, exactly top_k), undo class offsets, zero dead rows.
// ---------------------------------------------------------------------------
__global__ __launch_bounds__(512) void yolo_finalize(
    const float* __restrict__ merged, const float* __restrict__ conf_k,
    const float* __restrict__ cc_k, const float* __restrict__ cp_k,
    const int* __restrict__ keep, float* __restrict__ out)
{
    const int img = blockIdx.x;
    const int t   = threadIdx.x;
    const size_t base = (size_t)img * KTOP;

    __shared__ float s_c[KTOP];
    const int   kp = keep[base + t];
    const float co = kp ? conf_k[base + t] : -1.0f;
    s_c[t] = co;

    float* orow = out + base * 7;
    #pragma unroll
    for (int c = 0; c < 7; ++c) orow[(size_t)t * 7 + c] = 0.0f;
    __syncthreads();

    int rank = 0;
    for (int j = 0; j < KTOP; ++j) {
        const float v = s_c[j];
        rank += (v > co) || (v == co && j < t);
    }
    if (kp) {
        const float off = cp_k[base + t] * MAX_WH_F;
        float* d = orow + (size_t)rank * 7;
        d[0] = merged[(base + t) * 4 + 0] - off;
        d[1] = merged[(base + t) * 4 + 1];
        d[2] = merged[(base + t) * 4 + 2] - off;
        d[3] = merged[(base + t) * 4 + 3];
        d[4] = conf_k[base + t];
        d[5] = cc_k[base + t];
        d[6] = cp_k[base + t];
    }
}

// ---------------------------------------------------------------------------
extern "C" void kernel_launch(void* const* d_in, const int* in_sizes, int n_in,
                              void* d_out, int out_size, void* d_ws, size_t ws_size,
                              hipStream_t stream)
{
    (void)in_sizes; (void)n_in; (void)out_size; (void)ws_size;
    const float* pred = (const float*)d_in[0];
    float* out = (float*)d_out;

    char* ws = (char*)d_ws;
    size_t off = 0;
    auto carve = [&](size_t bytes) -> void* {
        void* p = ws + off;
        off += (bytes + 255) & ~(size_t)255;
        return p;
    };
    float* conf_k = (float*)carve((size_t)NB * KTOP * sizeof(float));
    float* cc_k   = (float*)carve((size_t)NB * KTOP * sizeof(float));
    float* cp_k   = (float*)carve((size_t)NB * KTOP * sizeof(float));
    float* a1     = (float*)carve((size_t)NB * KTOP * sizeof(float));
    int*   ok_k   = (int*)  carve((size_t)NB * KTOP * sizeof(int));
    int*   keep   = (int*)  carve((size_t)NB * KTOP * sizeof(int));
    float* obox   = (float*)carve((size_t)NB * KTOP * 4 * sizeof(float));
    float* merged = (float*)carve((size_t)NB * KTOP * 4 * sizeof(float));
    float* iou    = (float*)carve((size_t)NB * KTOP * KTOP * sizeof(float)); // 16 MB, L2-resident

    yolo_topk_gather<<<NB, 1024, 0, stream>>>(pred, conf_k, cc_k, cp_k, a1, ok_k, obox);
    yolo_iou_matrix<<<dim3(128, NB), 256, 0, stream>>>(obox, a1, iou);
    yolo_nms_scan<<<NB, 512, 0, stream>>>(obox, conf_k, ok_k, iou, keep, merged);
    yolo_finalize<<<NB, 512, 0, stream>>>(merged, conf_k, cc_k, cp_k, keep, out);
}